// QuadUnidirToDenseShift_79121887527417
// MI455X (gfx1250) — compile-verified
//
#include <hip/hip_runtime.h>

// ---- CDNA5 / gfx1250: wave32, WMMA f32 16x16x4 ----
typedef __attribute__((ext_vector_type(2))) float v2f;
typedef __attribute__((ext_vector_type(4))) float v4f;
typedef __attribute__((ext_vector_type(8))) float v8f;

#define DIM0 160
#define DIM1 192
#define DIM2 160
#define NVOX (DIM0 * DIM1 * DIM2)   // 4,915,200
#define NBATCH 8
#define VOX_PER_BLOCK 128           // 8 waves * 16 voxels
#define LDS_STRIDE 129              // pad: conflict-free WMMA->LDS scatter

__global__ __launch_bounds__(256)
void quad_dense_shift_kernel(const float* __restrict__ mat,  // [8][4][4]
                             float* __restrict__ out)        // [8][160][192][160][3]
{
    __shared__ float lds[NBATCH * LDS_STRIDE];   // [batch][voxel_local], padded

    const int t = threadIdx.x;
    const int l = t & 31;          // lane in wave32
    const int w = t >> 5;          // wave id 0..7
    const int vbase = blockIdx.x * VOX_PER_BLOCK;

    // ---------------- per-lane voxel homogeneous coordinate ----------------
    // A-matrix 16x4 f32 layout: lanes 0-15 -> M=lane, K=0..1 in VGPR0..1
    //                           lanes 16-31 -> M=lane-16, K=2..3 in VGPR0..1
    const int m = vbase + w * 16 + (l & 15);
    const int i   = m / (DIM1 * DIM2);
    const int rem = m - i * (DIM1 * DIM2);
    const int j   = rem / DIM2;
    const int k   = rem - j * DIM2;

    const float x0 = (float)i - 79.5f;   // (160-1)/2
    const float x1 = (float)j - 95.5f;   // (192-1)/2
    const float x2 = (float)k - 79.5f;   // (160-1)/2
    const float x3 = 1.0f;

    const bool lo16 = (l < 16);
    const float xa = lo16 ? x0 : x2;     // x[klo]
    const float xb = lo16 ? x1 : x3;     // x[klo+1]
    const int  klo = lo16 ? 0 : 2;

    // B-matrix 4x16 f32 layout: column N = lane&15; lanes 0-15 hold K=0..1,
    // lanes 16-31 hold K=2..3. Columns 8..15 duplicate batches 0..7 (unused).
    const int bcol = l & 15;
    const int be   = bcol & 7;           // keep loads in-bounds, EXEC all-1s
    const float* mb = mat + be * 16;     // M_be row-major [4][4]

    // ------------- q_vb = sum_c sum_k (x_c * x_k) * M_b[c][k] --------------
    // 4 accumulating WMMAs, K=4 each, cover all 16 monomials.
    const float xc[4] = {x0, x1, x2, x3};
    v8f acc = {};
#pragma unroll
    for (int c = 0; c < 4; ++c) {
        v2f a;
        a.x = xc[c] * xa;                // constant index: no select chains
        a.y = xc[c] * xb;
        v2f bm = *(const v2f*)(mb + c * 4 + klo);   // 8B-aligned
        acc = __builtin_amdgcn_wmma_f32_16x16x4_f32(
                  false, a, false, bm, (short)0, acc, false, false);
    }

    // ---------------- D -> LDS transpose: [batch][voxel_local] -------------
    // D layout: lane n (n = l&15) = column/batch; VGPR r holds voxel row
    //           r (lanes 0-15) or r+8 (lanes 16-31).
    const int hi = lo16 ? 0 : 8;
    if (bcol < 8) {
#pragma unroll
        for (int r = 0; r < 8; ++r) {
            lds[bcol * LDS_STRIDE + w * 16 + hi + r] = acc[r];
        }
    }
    __syncthreads();

    // -------- coalesced, 16B-aligned, non-temporal global stores -----------
    // One thread <-> (batch, 4-voxel group): 12 output floats = 3 float4s
    // with a compile-time zero pattern. Wave = one batch's contiguous 1536B
    // region (12 full cachelines), written by 3 b128 stores per lane.
    const int b = t >> 5;                // batch   0..7  (uniform per wave)
    const int g = t & 31;                // voxel quad id 0..31
    const float* lp = &lds[b * LDS_STRIDE + 4 * g];
    const float q0 = lp[0], q1 = lp[1], q2 = lp[2], q3 = lp[3];

    v4f s0; s0[0] = q0;   s0[1] = 0.0f; s0[2] = 0.0f; s0[3] = q1;
    v4f s1; s1[0] = 0.0f; s1[1] = 0.0f; s1[2] = q2;   s1[3] = 0.0f;
    v4f s2; s2[0] = 0.0f; s2[1] = q3;   s2[2] = 0.0f; s2[3] = 0.0f;

    const size_t off = ((size_t)b * NVOX + (size_t)(vbase + 4 * g)) * 3;
    __builtin_nontemporal_store(s0, (v4f*)(out + off));
    __builtin_nontemporal_store(s1, (v4f*)(out + off + 4));
    __builtin_nontemporal_store(s2, (v4f*)(out + off + 8));
}

extern "C" void kernel_launch(void* const* d_in, const int* in_sizes, int n_in,
                              void* d_out, int out_size, void* d_ws, size_t ws_size,
                              hipStream_t stream) {
    (void)in_sizes; (void)n_in; (void)out_size; (void)d_ws; (void)ws_size;
    const float* mat = (const float*)d_in[0];   // [8,4,4] float32
    float* out = (float*)d_out;                 // [8,160,192,160,3] float32

    const int blocks = NVOX / VOX_PER_BLOCK;    // 38,400 exactly, no tail
    quad_dense_shift_kernel<<<blocks, 256, 0, stream>>>(mat, out);
}